// SingleHeadAttention_76063870812190
// MI455X (gfx1250) — compile-verified
//
#include <hip/hip_runtime.h>
#include <hip/hip_bf16.h>
#include <math.h>

// ---------------------------------------------------------------------------
// Single-head causal attention, fp32, for MI455X (gfx1250).
//   B=4, S=4096, E=768, A=64.
// Memory traffic (~60 MB total) fits in 192MB L2 -> compute bound.
// Output is rounded to 1e-4, so we keep full fp32 precision and use
// V_WMMA_F32_16X16X4_F32 (fp32 matrix cores) for all three matmul stages.
// ---------------------------------------------------------------------------

typedef __attribute__((ext_vector_type(2))) float v2f;
typedef __attribute__((ext_vector_type(8))) float v8f;

#define EMBED 768
#define ADIM  64
#define SEQ   4096
#define BATCH 4
#define ROWS  (BATCH * SEQ)          // 16384 total rows
#define QTILES (ROWS / 16)           // 1024 16-row tiles
#define TILES_PER_BATCH (SEQ / 16)   // 256

// D = A(16x4 f32) * B(4x16 f32) + C(16x16 f32)
__device__ __forceinline__ v8f wmma4(v2f a, v2f b, v8f c) {
    // 8 args: (neg_a, A, neg_b, B, c_mod, C, reuse_a, reuse_b)
    return __builtin_amdgcn_wmma_f32_16x16x4_f32(
        false, a, false, b, (short)0, c, false, false);
}

// ---------------------------------------------------------------------------
// Kernel 1: Q/K/V projections.  grid = (QTILES/8, 3), block = 256 (8 waves).
// Each wave: 16 rows x 64 cols of one projection, K-loop over 768 in steps
// of 4 with fp32 WMMA.
// fp32 A layout (16x4): lane<16 holds row=lane, {V0=K0,V1=K1};
//                       lane>=16 holds row=lane-16, {V0=K2,V1=K3}.
// fp32 B layout (4x16): lane<16 col=lane {V0=K0,V1=K1}; lane>=16 {V0=K2,V1=K3}.
// fp32 C layout (16x16): VGPR r = row r (lanes 0-15) / row r+8 (lanes 16-31),
//                        col = lane&15.
// ---------------------------------------------------------------------------
__global__ __launch_bounds__(256)
void sha_proj_kernel(const float* __restrict__ X,
                     const float* __restrict__ Wq,
                     const float* __restrict__ Wk,
                     const float* __restrict__ Wv,
                     float* __restrict__ QKV) {
    const int wave = blockIdx.x * 8 + (threadIdx.x >> 5);   // 0..1023
    const int mat  = blockIdx.y;                            // 0=Q,1=K,2=V
    const float* __restrict__ W = (mat == 0) ? Wq : (mat == 1) ? Wk : Wv;
    float* __restrict__ O = QKV + (size_t)mat * ((size_t)ROWS * ADIM);

    const int lane = threadIdx.x & 31;
    const int lo = lane & 15;
    const int hi = lane >> 4;

    const size_t row0 = (size_t)wave * 16;
    const float* __restrict__ xr = X + (row0 + lo) * EMBED + 2 * hi;

    v8f acc[4] = {};
#pragma unroll 4
    for (int k = 0; k < EMBED; k += 4) {
        v2f a = *(const v2f*)(xr + k);                      // X[row][k..k+1] (+2 for hi half)
        const float* wk0 = W + (k + 2 * hi) * ADIM + lo;
#pragma unroll
        for (int t = 0; t < 4; ++t) {
            v2f b;
            b.x = wk0[t * 16];                              // W[k+2hi ][16t+lo]
            b.y = wk0[ADIM + t * 16];                       // W[k+2hi+1][16t+lo]
            acc[t] = wmma4(a, b, acc[t]);
        }
    }
#pragma unroll
    for (int t = 0; t < 4; ++t)
#pragma unroll
        for (int r = 0; r < 8; ++r)
            O[(row0 + r + 8 * hi) * ADIM + t * 16 + lo] = acc[t][r];
}

// ---------------------------------------------------------------------------
// Kernel 2: flash attention (causal, online softmax), fp32 WMMA.
// One wave per 16-row query tile.  grid = QTILES/8, block = 256 (8 waves).
// Probabilities are bounced through a per-wave LDS tile (stride 18 floats:
// even -> 8B-aligned b64 loads, and 18 is co-prime enough with 64 banks)
// to convert the WMMA C layout into the A layout for the P*V matmul.
// ---------------------------------------------------------------------------
__global__ __launch_bounds__(256)
void sha_attn_kernel(const float* __restrict__ QKV,
                     float* __restrict__ out) {
    const float* __restrict__ Q = QKV;
    const float* __restrict__ K = QKV + (size_t)ROWS * ADIM;
    const float* __restrict__ V = QKV + 2 * (size_t)ROWS * ADIM;

    __shared__ __align__(16) float Pbuf[8][16 * 18];

    const int wslot = threadIdx.x >> 5;
    const int wave  = blockIdx.x * 8 + wslot;               // 0..1023
    const int lane  = threadIdx.x & 31;
    const int lo = lane & 15;
    const int hi = lane >> 4;

    const int b  = wave / TILES_PER_BATCH;                  // batch
    const int qt = wave % TILES_PER_BATCH;                  // query tile in batch
    const size_t grow0 = (size_t)b * SEQ + (size_t)qt * 16; // global query row base

    // Preload Q tile (16x64) in fp32 A layout: 16 float2 operands.
    v2f aq[16];
    const float* __restrict__ qp = Q + (grow0 + lo) * ADIM + 2 * hi;
#pragma unroll
    for (int kk = 0; kk < 16; ++kk) aq[kk] = *(const v2f*)(qp + 4 * kk);

    v8f oacc[4] = {};
    float m[8], l[8];
#pragma unroll
    for (int r = 0; r < 8; ++r) { m[r] = -INFINITY; l[r] = 0.0f; }

    float* __restrict__ P = &Pbuf[wslot][0];
    const float invs = 0.125f;                              // 1/sqrt(64)
    const int nkb = qt + 1;                                 // causal: key blocks 0..qt

    for (int j = 0; j < nkb; ++j) {
        const size_t kb0 = (size_t)b * SEQ + (size_t)j * 16;

        if (j + 1 < nkb) {                                  // gfx1250 global_prefetch_b8
            __builtin_prefetch(K + (kb0 + 16 + lo) * ADIM, 0, 3);
            __builtin_prefetch(V + (kb0 + 16 + lo) * ADIM, 0, 3);
        }

        // ---- S = Q * K^T  (16x16, 16 fp32-WMMA k-steps) ----
        v8f s = {};
        const float* __restrict__ kp = K + (kb0 + lo) * ADIM + 2 * hi;
#pragma unroll
        for (int kk = 0; kk < 16; ++kk) {
            v2f bk = *(const v2f*)(kp + 4 * kk);            // K^T B-operand: contiguous pair
            s = wmma4(aq[kk], bk, s);
        }

        // ---- online softmax over this 16-key block ----
        const bool diag = (j == qt);
        float alpha[8];
#pragma unroll
        for (int r = 0; r < 8; ++r) {
            float sv = s[r] * invs;
            const int rq = r + 8 * hi;                      // row within tile
            if (diag) sv = (lo > rq) ? -INFINITY : sv;      // causal mask (select, no branch)

            // row max across the 16-lane half (wave32 shfl_xor, masks < 16)
            float bm = sv;
            bm = fmaxf(bm, __shfl_xor(bm, 1));
            bm = fmaxf(bm, __shfl_xor(bm, 2));
            bm = fmaxf(bm, __shfl_xor(bm, 4));
            bm = fmaxf(bm, __shfl_xor(bm, 8));
            const float mnew = fmaxf(m[r], bm);
            const float a = __expf(m[r] - mnew);            // exp(-inf)=0 on first block
            const float p = __expf(sv - mnew);

            float ps = p;
            ps += __shfl_xor(ps, 1);
            ps += __shfl_xor(ps, 2);
            ps += __shfl_xor(ps, 4);
            ps += __shfl_xor(ps, 8);

            l[r] = l[r] * a + ps;
            m[r] = mnew;
            alpha[r] = a;
            P[(r + 8 * hi) * 18 + lo] = p;                  // C layout -> LDS
        }

        // rescale running output by alpha (per row)
#pragma unroll
        for (int t = 0; t < 4; ++t)
#pragma unroll
            for (int r = 0; r < 8; ++r) oacc[t][r] *= alpha[r];

        // ---- O += P * V  (16x16 x 16x64, 16 fp32-WMMA ops) ----
#pragma unroll
        for (int kk = 0; kk < 16; kk += 4) {
            v2f ap = *(const v2f*)(P + lo * 18 + kk + 2 * hi);  // LDS -> A layout
            const float* __restrict__ vp = V + (kb0 + kk + 2 * hi) * ADIM + lo;
#pragma unroll
            for (int t = 0; t < 4; ++t) {
                v2f bv;
                bv.x = vp[t * 16];                          // V[kb0+kk+2hi ][16t+lo]
                bv.y = vp[ADIM + t * 16];                   // V[kb0+kk+2hi+1][16t+lo]
                oacc[t] = wmma4(ap, bv, oacc[t]);
            }
        }
    }

    // ---- epilogue: divide by softmax denominator, round like reference ----
    float rl[8];
#pragma unroll
    for (int r = 0; r < 8; ++r) rl[r] = 1.0f / l[r];
#pragma unroll
    for (int t = 0; t < 4; ++t)
#pragma unroll
        for (int r = 0; r < 8; ++r) {
            float v = oacc[t][r] * rl[r];
            v = rintf(v * 1.0e4f) * 1.0e-4f;                // torch.round(..., decimals=4)
            out[(grow0 + r + 8 * hi) * ADIM + t * 16 + lo] = v;
        }
}

// ---------------------------------------------------------------------------
extern "C" void kernel_launch(void* const* d_in, const int* in_sizes, int n_in,
                              void* d_out, int out_size, void* d_ws, size_t ws_size,
                              hipStream_t stream) {
    const float* X  = (const float*)d_in[0];   // [4,4096,768]
    const float* Wq = (const float*)d_in[1];   // [768,64]
    const float* Wk = (const float*)d_in[2];
    const float* Wv = (const float*)d_in[3];
    float* out = (float*)d_out;                // [4,4096,64]
    float* qkv = (float*)d_ws;                 // 3 * 16384 * 64 floats = 12 MB

    dim3 blk(256);
    dim3 gridProj(QTILES / 8, 3);              // 1024 waves per projection
    sha_proj_kernel<<<gridProj, blk, 0, stream>>>(X, Wq, Wk, Wv, qkv);

    dim3 gridAttn(QTILES / 8);                 // 1024 query-tile waves
    sha_attn_kernel<<<gridAttn, blk, 0, stream>>>(qkv, out);
}